// AnalyticalMNet_15994458211298
// MI455X (gfx1250) — compile-verified
//
#include <hip/hip_runtime.h>

#define HW        (512 * 512)
#define NBATCH    64
#define WGS_PER_B 32
#define CHUNK     (HW / WGS_PER_B)   // 8192 pixels per workgroup
#define TPB       256

typedef __attribute__((ext_vector_type(2))) float v2f;
typedef __attribute__((ext_vector_type(8))) float v8f;

// ---------------------------------------------------------------------------
// Stage 1: bandwidth-bound streaming. grid = (32 WGs, 64 batches), 256 thr.
// Each thread handles 32 pixels (8 iters x 4 px) with pure b128 loads.
// Deterministic in-block reduction (shfl_xor + LDS), writes padded 16-float
// partial vector per (batch, wg) into workspace. No atomics.
// ---------------------------------------------------------------------------
__global__ __launch_bounds__(TPB) void stain_partials(
    const float* __restrict__ Y,   // (B, HW, 3)
    const float* __restrict__ C,   // (B, 2, HW)
    float* __restrict__ partials)  // (B, 32, 16)
{
    const int wg  = blockIdx.x;    // 0..31
    const int b   = blockIdx.y;    // 0..63
    const int tid = threadIdx.x;

    const float* c0 = C + (size_t)b * 2 * HW;
    const float* c1 = c0 + HW;
    const float* y  = Y + (size_t)b * HW * 3;

    const int base = wg * CHUNK;

    float g00 = 0.f, g01 = 0.f, g11 = 0.f;
    float yc00 = 0.f, yc01 = 0.f, yc10 = 0.f, yc11 = 0.f, yc20 = 0.f, yc21 = 0.f;

#pragma unroll
    for (int it = 0; it < CHUNK / (TPB * 4); ++it) {
        const int p = base + (it * TPB + tid) * 4;
        const float4 a  = *(const float4*)(c0 + p);        // stain0, 4 px
        const float4 c  = *(const float4*)(c1 + p);        // stain1, 4 px
        const float4 q0 = *(const float4*)(y + 3 * p);     // 12 interleaved y's
        const float4 q1 = *(const float4*)(y + 3 * p + 4);
        const float4 q2 = *(const float4*)(y + 3 * p + 8);

        // gram
        g00 += a.x * a.x + a.y * a.y + a.z * a.z + a.w * a.w;
        g01 += a.x * c.x + a.y * c.y + a.z * c.z + a.w * c.w;
        g11 += c.x * c.x + c.y * c.y + c.z * c.z + c.w * c.w;

        // y per pixel: px0={q0.x,q0.y,q0.z} px1={q0.w,q1.x,q1.y}
        //              px2={q1.z,q1.w,q2.x} px3={q2.y,q2.z,q2.w}
        yc00 += q0.x * a.x + q0.w * a.y + q1.z * a.z + q2.y * a.w;
        yc01 += q0.x * c.x + q0.w * c.y + q1.z * c.z + q2.y * c.w;
        yc10 += q0.y * a.x + q1.x * a.y + q1.w * a.z + q2.z * a.w;
        yc11 += q0.y * c.x + q1.x * c.y + q1.w * c.z + q2.z * c.w;
        yc20 += q0.z * a.x + q1.y * a.y + q2.x * a.z + q2.w * a.w;
        yc21 += q0.z * c.x + q1.y * c.y + q2.x * c.z + q2.w * c.w;
    }

    float acc[9] = {yc00, yc01, yc10, yc11, yc20, yc21, g00, g01, g11};

    // wave32 butterfly reduction
#pragma unroll
    for (int off = 16; off >= 1; off >>= 1) {
#pragma unroll
        for (int v = 0; v < 9; ++v)
            acc[v] += __shfl_xor(acc[v], off, 32);
    }

    __shared__ float red[TPB / 32][16];
    const int wave = tid >> 5;
    const int lane = tid & 31;
    if (lane == 0) {
#pragma unroll
        for (int v = 0; v < 9; ++v) red[wave][v] = acc[v];
#pragma unroll
        for (int v = 9; v < 16; ++v) red[wave][v] = 0.f;
    }
    __syncthreads();
    if (tid < 16) {
        float s = 0.f;
#pragma unroll
        for (int w = 0; w < TPB / 32; ++w) s += red[w][tid];
        partials[((size_t)b * WGS_PER_B + wg) * 16 + tid] = s;
    }
}

// ---------------------------------------------------------------------------
// Stage 2: one wave (32 thr) per batch. Reduce the 32 partial vectors with
// 8 chained V_WMMA_F32_16X16X4_F32: A = partials (rows = value index,
// K = wg index; ISA f32 16x4 layout: lanes 0-15 -> K=0,1 ; lanes 16-31 ->
// K=2,3), B = all-ones => D[v][n] = sum_k A[v][k], accumulated over chunks.
// Lane 0 then performs the closed-form 2x2 inverse + 3x2 solve.
// ---------------------------------------------------------------------------
__global__ __launch_bounds__(32) void stain_solve(
    const float* __restrict__ partials,  // (B, 32, 16)
    const float* __restrict__ M_ref,     // (B, 3, 2)
    const float* __restrict__ sigma_sq,  // scalar
    const float* __restrict__ lambda_sq, // scalar
    float* __restrict__ out)             // [B*6 mean | B*4 var]
{
    const int b    = blockIdx.x;
    const int lane = threadIdx.x;
    const int m    = lane & 15;            // matrix row = value index
    const int kofs = (lane >> 4) << 1;     // 0 for lanes 0-15, 2 for 16-31
    const float* base = partials + (size_t)b * WGS_PER_B * 16;

    v8f acc = {};
    v2f ones;
    ones.x = 1.f;
    ones.y = 1.f;

#pragma unroll
    for (int chunk = 0; chunk < WGS_PER_B / 4; ++chunk) {
        const int k = chunk * 4 + kofs;
        v2f a;
        a.x = base[(size_t)k * 16 + m];        // A[m][k]   = partials[k][m]
        a.y = base[(size_t)(k + 1) * 16 + m];  // A[m][k+1]
        acc = __builtin_amdgcn_wmma_f32_16x16x4_f32(
            /*neg_a=*/false, a, /*neg_b=*/false, ones,
            /*c_mod=*/(short)0, acc, /*reuse_a=*/false, /*reuse_b=*/false);
    }

    // D layout: lane n (n<16), vgpr r -> D[r][n]; lane 16+n, vgpr r -> D[8+r][n].
    // All columns identical (B == ones). total[v<=7] = acc[v] @ lane 0,
    // total[8] = acc[0] @ lane 16.
    const float g11 = __shfl(acc[0], 16, 32);

    if (lane == 0) {
        const float inv_s = 1.f / sigma_sq[0];
        const float inv_l = 1.f / lambda_sq[0];
        const float yc_[6] = {acc[0], acc[1], acc[2], acc[3], acc[4], acc[5]};
        const float g00 = acc[6], g01 = acc[7];

        const float A00 = inv_s + inv_l * g00;
        const float A01 = inv_l * g01;
        const float A11 = inv_s + inv_l * g11;
        const float invdet = 1.f / (A00 * A11 - A01 * A01);
        const float V00 = A11 * invdet;
        const float V01 = -A01 * invdet;
        const float V11 = A00 * invdet;

        const float* M  = M_ref + b * 6;
        float*       om = out + b * 6;
#pragma unroll
        for (int s = 0; s < 3; ++s) {
            const float t0 = inv_s * M[s * 2 + 0] + inv_l * yc_[s * 2 + 0];
            const float t1 = inv_s * M[s * 2 + 1] + inv_l * yc_[s * 2 + 1];
            om[s * 2 + 0] = t0 * V00 + t1 * V01;
            om[s * 2 + 1] = t0 * V01 + t1 * V11;
        }
        float* ov = out + NBATCH * 6 + b * 4;
        ov[0] = V00;
        ov[1] = V01;
        ov[2] = V01;
        ov[3] = V11;
    }
}

extern "C" void kernel_launch(void* const* d_in, const int* in_sizes, int n_in,
                              void* d_out, int out_size, void* d_ws, size_t ws_size,
                              hipStream_t stream) {
    const float* Y   = (const float*)d_in[0];  // Y_OD   (B, HW, 3)
    const float* C   = (const float*)d_in[1];  // C_mean (B, 2, H, W)
    const float* M   = (const float*)d_in[2];  // M_ref  (B, 3, 2)
    const float* sig = (const float*)d_in[3];  // sigma_sq
    const float* lam = (const float*)d_in[4];  // lambda_sq
    float* out      = (float*)d_out;
    float* partials = (float*)d_ws;            // 64*32*16 floats = 128 KB

    dim3 grid1(WGS_PER_B, NBATCH);
    stain_partials<<<grid1, TPB, 0, stream>>>(Y, C, partials);
    stain_solve<<<NBATCH, 32, 0, stream>>>(partials, M, sig, lam, out);
}